// Normalize_66657892434063
// MI455X (gfx1250) — compile-verified
//
#include <hip/hip_runtime.h>
#include <stdint.h>

// GCN adjacency normalization: out = D^-1/2 * (A with diag := atom_mask) * D^-1/2
// B=256 batches of 512x512 f32. Pure bandwidth problem (~0.2 GFLOP vs 512 MB
// traffic): target is the 23.3 TB/s HBM roofline with L2 reuse between passes.
// CDNA5 paths used: global_load_async_to_lds_b128 (+ s_wait_asynccnt pipelining)
// for the reduction pass, wave32 shuffle reduce, NT temporal hints for the
// streaming pass.

#define NDIM 512
#define WAVES 16
#define TPB (WAVES * 32)
#define ROWS_PER_WAVE (NDIM / WAVES)  // 32

// clang native vector: accepted by __builtin_nontemporal_{load,store},
// lowers to b128 global/ds ops.
typedef float v4f __attribute__((ext_vector_type(4)));

// One row (2 KB) global -> LDS, 4 x b128-per-lane async copies (512 B each).
// Per CDNA5 ISA: INST_OFFSET is added to BOTH the LDS and the
// global address, so one (lds,gaddr) pair + immediate offsets covers the row.
__device__ __forceinline__ void issue_row_async(uint32_t lds_byte_addr, uint64_t gaddr) {
  asm volatile(
      "global_load_async_to_lds_b128 %0, %1, off\n\t"
      "global_load_async_to_lds_b128 %0, %1, off offset:512\n\t"
      "global_load_async_to_lds_b128 %0, %1, off offset:1024\n\t"
      "global_load_async_to_lds_b128 %0, %1, off offset:1536"
      :
      : "v"(lds_byte_addr), "v"(gaddr)
      : "memory");
}

__global__ __launch_bounds__(TPB) void gcn_norm_kernel(const float* __restrict__ adj,
                                                       float* __restrict__ out) {
  __shared__ __align__(16) float s_dinv[NDIM];                 // 2 KB
  __shared__ __align__(16) float s_stage[WAVES][2][NDIM];      // 64 KB double buffer

  const int b    = blockIdx.x;
  const int tid  = threadIdx.x;
  const int wave = tid >> 5;
  const int lane = tid & 31;

  const float* A = adj + (size_t)b * NDIM * NDIM;
  float*       O = out + (size_t)b * NDIM * NDIM;

  // ---------------- Pass 1: degrees via async-staged row reductions ----------
  {
    // LDS generic-pointer low 32 bits == LDS byte offset (aperture in high bits)
    const uint32_t lds0 = (uint32_t)(uintptr_t)&s_stage[wave][0][lane * 4];
    const uint32_t lds1 = (uint32_t)(uintptr_t)&s_stage[wave][1][lane * 4];

    // prologue: row 'wave' into buffer 0  (ASYNCcnt = 4)
    issue_row_async(lds0, (uint64_t)(uintptr_t)(A + (size_t)wave * NDIM + lane * 4));

    for (int k = 0; k < ROWS_PER_WAVE; ++k) {
      const int row = wave + k * WAVES;
      if (k + 1 < ROWS_PER_WAVE) {
        const int nrow = wave + (k + 1) * WAVES;
        issue_row_async(((k + 1) & 1) ? lds1 : lds0,
                        (uint64_t)(uintptr_t)(A + (size_t)nrow * NDIM + lane * 4));
        // 4 just-issued copies may stay in flight; current buffer's 4 are done
        asm volatile("s_wait_asynccnt 4" ::: "memory");
      } else {
        asm volatile("s_wait_asynccnt 0" ::: "memory");
      }

      // 16 floats per lane, then wave32 butterfly reduce
      const v4f* src = (const v4f*)&s_stage[wave][k & 1][0];
      float s = 0.0f;
#pragma unroll
      for (int i = 0; i < 4; ++i) {
        v4f v = src[lane + 32 * i];
        s += (v.x + v.y) + (v.z + v.w);
      }
      s += __shfl_xor(s, 16, 32);
      s += __shfl_xor(s, 8, 32);
      s += __shfl_xor(s, 4, 32);
      s += __shfl_xor(s, 2, 32);
      s += __shfl_xor(s, 1, 32);

      const float diag = s_stage[wave][k & 1][row];   // original A[row][row]
      const float mask = (s != 0.0f) ? 1.0f : 0.0f;   // atom_mask
      const float deg  = (s - diag) + mask;           // degree of A_hat row
      const float dv   = (deg > 0.0f) ? (1.0f / sqrtf(deg)) : 0.0f;
      if (lane == 0) s_dinv[row] = dv;
    }
  }
  __syncthreads();

  // ---------------- Pass 2: out[i][j] = dinv[i]*dinv[j]*A_hat[i][j] ----------
  // Re-read is an L2 hit (1 MB tile resident from pass 1). NT loads: last use,
  // don't promote. NT stores: stream 256 MB output past L2.
  const int totalChunks = NDIM * (NDIM / 4);  // 65536 float4 per batch
  for (int c = tid; c < totalChunks; c += TPB) {
    const int i  = c >> 7;            // 128 float4 chunks per row
    const int j0 = (c & 127) << 2;

    v4f a = __builtin_nontemporal_load((const v4f*)(A + (size_t)i * NDIM + j0));
    const float di = s_dinv[i];
    const v4f   dj = *(const v4f*)&s_dinv[j0];

    // diagonal of A_hat is atom_mask[i]; mask[i] == (dinv[i] != 0)
    const float m = (di != 0.0f) ? 1.0f : 0.0f;
    a.x = (i == j0 + 0) ? m : a.x;
    a.y = (i == j0 + 1) ? m : a.y;
    a.z = (i == j0 + 2) ? m : a.z;
    a.w = (i == j0 + 3) ? m : a.w;

    v4f o;
    o.x = di * dj.x * a.x;
    o.y = di * dj.y * a.y;
    o.z = di * dj.z * a.z;
    o.w = di * dj.w * a.w;
    __builtin_nontemporal_store(o, (v4f*)(O + (size_t)i * NDIM + j0));
  }
}

extern "C" void kernel_launch(void* const* d_in, const int* in_sizes, int n_in,
                              void* d_out, int out_size, void* d_ws, size_t ws_size,
                              hipStream_t stream) {
  (void)n_in; (void)out_size; (void)d_ws; (void)ws_size;
  const float* adj = (const float*)d_in[0];
  float* out = (float*)d_out;
  const int B = in_sizes[0] / (NDIM * NDIM);  // 256 for the reference shapes
  gcn_norm_kernel<<<dim3(B), dim3(TPB), 0, stream>>>(adj, out);
}